// SuperGAT_54881092108451
// MI455X (gfx1250) — compile-verified
//
#include <hip/hip_runtime.h>
#include <hip/hip_bf16.h>
#include <cstdint>
#include <cstddef>

// SuperGAT-MX, 2 layers. N=100000, E=1.6M(+N self loops), IN=512, H=8, HID=8, OUT=16.
#define NNODES 100000
#define HEADS 8

typedef __attribute__((ext_vector_type(2))) float v2f;
typedef __attribute__((ext_vector_type(8))) float v8f;

// ---------------------------------------------------------------------------
// GEMM via V_WMMA_F32_16X16X4_F32 (fp32-exact matrix cores, CDNA5 wave32).
// Y[nrows x NCOL] = X[nrows x KDIM] * W[KDIM x NCOL]
// Block = 128 threads (4 waves). Each block: 64 rows; each wave: 16 rows x NCOL.
//
// LDS layouts:
//  - X tile 64x64, row stride 68 dwords: column reads land on distinct banks.
//  - W tile stored K-pair interleaved: Ws[kpair][n][2] with kpair stride
//    NCOL*2+32 dwords (== 32 mod 64), so each B fragment is ONE aligned
//    ds_load_b64 and lanes 0-15 / 16-31 use disjoint bank halves (0-31 / 32-63).
//    This removes the v_dual_mov repacking seen in front of every WMMA.
// ---------------------------------------------------------------------------
template<int KDIM, int NCOL>
__global__ __launch_bounds__(128)
void gemm_wmma_f32(const float* __restrict__ X, const float* __restrict__ W,
                   float* __restrict__ Y, int nrows) {
  constexpr int XS  = 68;             // padded LDS stride (dwords) for X tile
  constexpr int WSP = NCOL * 2 + 32;  // dword stride per K-pair row of W tile
  constexpr int NT  = NCOL / 16;      // WMMA tiles along N
  __shared__ float Xs[64 * XS];
  __shared__ float Ws[32 * WSP];

  const int tid  = threadIdx.x;
  const int lane = tid & 31;
  const int wave = tid >> 5;            // 0..3 -> 16-row strip
  const int m    = lane & 15;           // M (A) / N (B) index within tile
  const int kp   = (lane >> 4) << 1;    // K pair: lanes 0-15 -> K0/K1, 16-31 -> K2/K3
  const int hw   = lane >> 4;           // half-wave select
  const int rowBase = blockIdx.x * 64;

  v8f acc[NT];
  #pragma unroll
  for (int j = 0; j < NT; ++j) acc[j] = (v8f){};

  for (int kb = 0; kb < KDIM; kb += 64) {
    // --- stage X tile 64x64 (coalesced float4, row-clamped for last block) ---
    {
      const int r0 = tid >> 4;            // 0..7
      const int c  = (tid & 15) * 4;      // 0..60
      #pragma unroll
      for (int i = 0; i < 8; ++i) {
        int r  = r0 + i * 8;
        int gr = rowBase + r; if (gr >= nrows) gr = nrows - 1;
        const float4 v = *reinterpret_cast<const float4*>(X + (size_t)gr * KDIM + kb + c);
        float* dst = Xs + r * XS + c;
        dst[0] = v.x; dst[1] = v.y; dst[2] = v.z; dst[3] = v.w;
      }
    }
    // --- stage W tile 64xNCOL into K-pair interleaved layout ---
    {
      constexpr int Q = 64 * NCOL / 4;
      for (int idx = tid; idx < Q; idx += 128) {
        const int r = idx / (NCOL / 4);
        const int c = (idx % (NCOL / 4)) * 4;
        const float4 v = *reinterpret_cast<const float4*>(W + (size_t)(kb + r) * NCOL + c);
        float* dst = Ws + (r >> 1) * WSP + (r & 1);   // element stride 2 within row
        dst[(c + 0) * 2] = v.x;
        dst[(c + 1) * 2] = v.y;
        dst[(c + 2) * 2] = v.z;
        dst[(c + 3) * 2] = v.w;
      }
    }
    __syncthreads();

    // --- MAC loop: 16 WMMA K-steps of 4; EXEC is all-ones here ---
    #pragma unroll
    for (int kk = 0; kk < 64; kk += 4) {
      const float* xa = Xs + (wave * 16 + m) * XS + kk + kp;
      v2f a; a.x = xa[0]; a.y = xa[1];                 // ds_load_b64 (A pair)
      const float* wrow = Ws + ((kk >> 1) + hw) * WSP + m * 2;
      #pragma unroll
      for (int j = 0; j < NT; ++j) {
        const float* wb = wrow + j * 32;               // (j*16+m)*2
        v2f b; b.x = wb[0]; b.y = wb[1];               // single aligned ds_load_b64
        acc[j] = __builtin_amdgcn_wmma_f32_16x16x4_f32(
            /*neg_a=*/false, a, /*neg_b=*/false, b,
            /*c_mod=*/(short)0, acc[j], /*reuse_a=*/false, /*reuse_b=*/false);
      }
    }
    __syncthreads();
  }

  // --- store: VGPR v holds M = v + 8*(lane>>4), N = lane&15 ---
  #pragma unroll
  for (int j = 0; j < NT; ++j) {
    #pragma unroll
    for (int v = 0; v < 8; ++v) {
      const int row = rowBase + wave * 16 + hw * 8 + v;
      if (row < nrows) Y[(size_t)row * NCOL + j * 16 + m] = acc[j][v];
    }
  }
}

// ---------------------------------------------------------------------------
// Edge-phase helpers
// ---------------------------------------------------------------------------
__device__ __forceinline__ void edge_sd(const long long* __restrict__ ei,
                                        long long E, long long e, int& s, int& d) {
  if (e < E) { s = (int)ei[e]; d = (int)ei[E + e]; }
  else       { s = d = (int)(e - E); }   // appended self-loops
}

__device__ __forceinline__ void atomicMaxF(float* addr, float val) {
  int old = __float_as_int(*addr);
  while (__int_as_float(old) < val) {
    const int prev = atomicCAS(reinterpret_cast<int*>(addr), old, __float_as_int(val));
    if (prev == old) break;
    old = prev;
  }
}

// Pass 1: one thread per (edge, head). MX gate + leaky relu; seg-max via atomics.
template<int C>
__global__ __launch_bounds__(256)
void edge_alpha_kernel(const float* __restrict__ hfeat,
                       const long long* __restrict__ ei, long long E, long long Etot,
                       const float* __restrict__ att_l, const float* __restrict__ att_r,
                       float* __restrict__ alpha_buf, float* __restrict__ amax) {
  const long long gid = (long long)blockIdx.x * blockDim.x + threadIdx.x;
  const long long e = gid >> 3;
  const int h = (int)(gid & 7);
  if (e >= Etot) return;
  int s, d; edge_sd(ei, E, e, s, d);
  const float* hj = hfeat + ((size_t)s * HEADS + h) * C;  // source feats
  const float* hi = hfeat + ((size_t)d * HEADS + h) * C;  // dest feats
  const float* al = att_l + h * C;
  const float* ar = att_r + h * C;
  float logit = 0.f, sl = 0.f, sr = 0.f;
  #pragma unroll
  for (int c = 0; c < C; c += 4) {
    const float4 xj = *reinterpret_cast<const float4*>(hj + c);
    const float4 xi = *reinterpret_cast<const float4*>(hi + c);
    const float4 wl = *reinterpret_cast<const float4*>(al + c);
    const float4 wr = *reinterpret_cast<const float4*>(ar + c);
    logit += xi.x * xj.x + xi.y * xj.y + xi.z * xj.z + xi.w * xj.w;
    sl    += xj.x * wl.x + xj.y * wl.y + xj.z * wl.z + xj.w * wl.w;
    sr    += xi.x * wr.x + xi.y * wr.y + xi.z * wr.z + xi.w * wr.w;
  }
  float a = (sl + sr) * (1.0f / (1.0f + __expf(-logit)));  // sigmoid MX gate
  a = a > 0.f ? a : 0.2f * a;                              // leaky relu
  alpha_buf[e * HEADS + h] = a;
  atomicMaxF(&amax[(size_t)d * HEADS + h], a);
}

// Pass 2: exp(alpha - max), accumulate denom and weighted source feats (L2-resident atomics).
template<int C>
__global__ __launch_bounds__(256)
void edge_agg_kernel(const float* __restrict__ hfeat,
                     const long long* __restrict__ ei, long long E, long long Etot,
                     const float* __restrict__ alpha_buf, const float* __restrict__ amax,
                     float* __restrict__ denom, float* __restrict__ agg) {
  const long long gid = (long long)blockIdx.x * blockDim.x + threadIdx.x;
  const long long e = gid >> 3;
  const int h = (int)(gid & 7);
  if (e >= Etot) return;
  int s, d; edge_sd(ei, E, e, s, d);
  const float w = __expf(alpha_buf[e * HEADS + h] - amax[(size_t)d * HEADS + h]);
  atomicAdd(&denom[(size_t)d * HEADS + h], w);
  const float* hj = hfeat + ((size_t)s * HEADS + h) * C;
  float* o = agg + ((size_t)d * HEADS + h) * C;
  #pragma unroll
  for (int c = 0; c < C; ++c) atomicAdd(&o[c], w * hj[c]);
}

// Layer-1 epilogue: normalize, +bias, ELU (in place -> becomes GEMM2 input).
__global__ __launch_bounds__(256)
void finalize1_kernel(float* __restrict__ agg, const float* __restrict__ denom,
                      const float* __restrict__ b, int n) {
  const int gid = blockIdx.x * blockDim.x + threadIdx.x;
  if (gid >= n * 64) return;
  const int node = gid >> 6;
  const int hc = gid & 63;
  const float v = agg[gid] / (denom[node * 8 + (hc >> 3)] + 1e-16f) + b[hc];
  agg[gid] = v > 0.f ? v : (__expf(v) - 1.f);   // ELU
}

// Layer-2 epilogue: normalize, mean over heads, +bias, log_softmax over 16 ch.
__global__ __launch_bounds__(256)
void finalize2_kernel(const float* __restrict__ agg, const float* __restrict__ denom,
                      const float* __restrict__ b, float* __restrict__ out, int n) {
  const int node = blockIdx.x * blockDim.x + threadIdx.x;
  if (node >= n) return;
  float dn[8];
  #pragma unroll
  for (int h = 0; h < 8; ++h) dn[h] = 1.0f / (denom[(size_t)node * 8 + h] + 1e-16f);
  float v[16];
  #pragma unroll
  for (int c = 0; c < 16; ++c) {
    float s = 0.f;
    #pragma unroll
    for (int h = 0; h < 8; ++h) s += agg[((size_t)node * 8 + h) * 16 + c] * dn[h];
    v[c] = s * 0.125f + b[c];
  }
  float m = v[0];
  #pragma unroll
  for (int c = 1; c < 16; ++c) m = fmaxf(m, v[c]);
  float lse = 0.f;
  #pragma unroll
  for (int c = 0; c < 16; ++c) lse += __expf(v[c] - m);
  lse = __logf(lse);
  #pragma unroll
  for (int c = 0; c < 16; ++c) out[(size_t)node * 16 + c] = v[c] - m - lse;
}

__global__ __launch_bounds__(256)
void fill_kernel(float* __restrict__ p, float v, size_t n) {
  const size_t i = (size_t)blockIdx.x * blockDim.x + threadIdx.x;
  if (i < n) p[i] = v;
}

// ---------------------------------------------------------------------------
extern "C" void kernel_launch(void* const* d_in, const int* in_sizes, int n_in,
                              void* d_out, int out_size, void* d_ws, size_t ws_size,
                              hipStream_t stream) {
  const float*     x     = (const float*)d_in[0];
  const long long* ei    = (const long long*)d_in[1];   // int64 (2,E) row-major
  const float*     W1    = (const float*)d_in[2];
  const float*     attl1 = (const float*)d_in[3];
  const float*     attr1 = (const float*)d_in[4];
  const float*     b1    = (const float*)d_in[5];
  const float*     W2    = (const float*)d_in[6];
  const float*     attl2 = (const float*)d_in[7];
  const float*     attr2 = (const float*)d_in[8];
  const float*     b2    = (const float*)d_in[9];
  float*           out   = (float*)d_out;

  const int n = NNODES;
  const long long E = (long long)in_sizes[1] / 2;
  const long long Etot = E + n;

  // workspace layout (floats)
  float* ws = (float*)d_ws;
  float* h1    = ws;              size_t o = (size_t)n * 64;
  float* agg1  = ws + o;          o += (size_t)n * 64;
  float* h2    = ws + o;          o += (size_t)n * 128;
  float* agg2  = ws + o;          o += (size_t)n * 128;
  float* amax  = ws + o;          o += (size_t)n * 8;
  float* denom = ws + o;          o += (size_t)n * 8;
  float* alpha = ws + o;          // Etot*8

  const int   gemmGrid = (n + 63) / 64;
  const long long eth  = Etot * 8;
  const int   eGrid    = (int)((eth + 255) / 256);
  const float ninf     = -__builtin_inff();

  // ---- layer 1 ----
  gemm_wmma_f32<512, 64><<<gemmGrid, 128, 0, stream>>>(x, W1, h1, n);
  fill_kernel<<<(int)(((size_t)n * 8  + 255) / 256), 256, 0, stream>>>(amax,  ninf, (size_t)n * 8);
  fill_kernel<<<(int)(((size_t)n * 8  + 255) / 256), 256, 0, stream>>>(denom, 0.f,  (size_t)n * 8);
  fill_kernel<<<(int)(((size_t)n * 64 + 255) / 256), 256, 0, stream>>>(agg1,  0.f,  (size_t)n * 64);
  edge_alpha_kernel<8><<<eGrid, 256, 0, stream>>>(h1, ei, E, Etot, attl1, attr1, alpha, amax);
  edge_agg_kernel<8><<<eGrid, 256, 0, stream>>>(h1, ei, E, Etot, alpha, amax, denom, agg1);
  finalize1_kernel<<<(n * 64 + 255) / 256, 256, 0, stream>>>(agg1, denom, b1, n);

  // ---- layer 2 ----
  gemm_wmma_f32<64, 128><<<gemmGrid, 128, 0, stream>>>(agg1, W2, h2, n);
  fill_kernel<<<(int)(((size_t)n * 8   + 255) / 256), 256, 0, stream>>>(amax,  ninf, (size_t)n * 8);
  fill_kernel<<<(int)(((size_t)n * 8   + 255) / 256), 256, 0, stream>>>(denom, 0.f,  (size_t)n * 8);
  fill_kernel<<<(int)(((size_t)n * 128 + 255) / 256), 256, 0, stream>>>(agg2,  0.f,  (size_t)n * 128);
  edge_alpha_kernel<16><<<eGrid, 256, 0, stream>>>(h2, ei, E, Etot, attl2, attr2, alpha, amax);
  edge_agg_kernel<16><<<eGrid, 256, 0, stream>>>(h2, ei, E, Etot, alpha, amax, denom, agg2);
  finalize2_kernel<<<(n + 255) / 256, 256, 0, stream>>>(agg2, denom, b2, out, n);
}